// compute_similarity_83640193122534
// MI455X (gfx1250) — compile-verified
//
#include <hip/hip_runtime.h>

typedef __attribute__((ext_vector_type(2))) float v2f;
typedef __attribute__((ext_vector_type(8))) float v8f;

#define NMAPS   8
#define BATCH   32
#define DDIM    (3 * 224 * 224)      // 150528
#define CHUNK   1568                 // floats of D per wave; 1568/32 = 49 unrolled iters
#define NCHUNKS (DDIM / CHUNK)       // 96
#define NPAIRS  (BATCH / 2)          // 16 batch pairs (b0=2p, b1=2p+1)
#define WAVES_PER_BLOCK 8
#define CGROUPS (NCHUNKS / WAVES_PER_BLOCK)   // 12 chunk-groups per pair
#define EPS     1e-8f

// ---------------------------------------------------------------------------
// Zero the 8x8x32 Gram scratch (graph-capture-safe, no hipMemset*).
// ---------------------------------------------------------------------------
__global__ void zero_ws_kernel(float* __restrict__ ws) {
    int i = blockIdx.x * blockDim.x + threadIdx.x;
    if (i < NMAPS * NMAPS * BATCH) ws[i] = 0.0f;
}

// ---------------------------------------------------------------------------
// One block = one batch pair (b0,b1) x one chunk-group; its 8 waves take 8
// consecutive D-chunks. 16x16 WMMA tile: rows 0-7 = maps of b0, rows 8-15 =
// maps of b1; D = A*A^T puts both 8x8 Gram blocks on the block diagonal.
//
// f32 16x16x4 operand layouts (lane L):
//   A: M = L%16, K = 2*(L/16) + vgpr      B: N = L%16, K = 2*(L/16) + vgpr
// identical register images for a symmetric Gram -> one float2 load per lane
// per K=4 step serves as both A and B.
//
// Waves reduce their Gram blocks into LDS (ds_add_f32), then the block does
// 128 global fp32 atomics -> 24.5K global atomics total instead of 200K.
// ---------------------------------------------------------------------------
__global__ __launch_bounds__(256) void gram_wmma_kernel(
        const float* __restrict__ x, float* __restrict__ ws) {
    __shared__ float g[2 * NMAPS * NMAPS];   // [2][8][8] Gram accumulators

    const int wave = threadIdx.x >> 5;
    const int lane = threadIdx.x & 31;
    const int pair = blockIdx.x % NPAIRS;
    const int cg   = blockIdx.x / NPAIRS;    // 0..CGROUPS-1
    const int chunk = cg * WAVES_PER_BLOCK + wave;
    const int b0 = pair * 2;
    const int b1 = b0 + 1;

    if (threadIdx.x < 2 * NMAPS * NMAPS) g[threadIdx.x] = 0.0f;
    __syncthreads();

    const int m   = lane & 15;              // tile row (A) == tile col (B)
    const int kb  = (lane >> 4) * 2;        // K sub-offset for this half-wave
    const int row = m & 7;                  // map index 0..7
    const int b   = (m < 8) ? b0 : b1;      // which batch element this row is

    const float* p = x + ((size_t)row * BATCH + b) * DDIM
                       + (size_t)chunk * CHUNK + kb;

    v8f c0 = {};
    v8f c1 = {};
    #pragma unroll 4
    for (int kk = 0; kk < CHUNK; kk += 8) {
        v2f a0 = *(const v2f*)(p + kk);       // K = kk .. kk+3   (this lane's share)
        v2f a1 = *(const v2f*)(p + kk + 4);   // K = kk+4 .. kk+7
        // D = A * A^T + C   (v_wmma_f32_16x16x4_f32)
        c0 = __builtin_amdgcn_wmma_f32_16x16x4_f32(
                false, a0, false, a0, (short)0, c0, false, false);
        c1 = __builtin_amdgcn_wmma_f32_16x16x4_f32(
                false, a1, false, a1, (short)0, c1, false, false);
    }
    v8f c = c0 + c1;

    // C/D layout: lanes 0-15 -> N=lane, M=r ; lanes 16-31 -> N=lane-16, M=r+8.
    // b0 diag block (M<8, N<8): lanes 0-7.  b1 diag block (M>=8, N>=8): lanes 24-31.
    if (lane < 8) {
        #pragma unroll
        for (int r = 0; r < 8; ++r)
            atomicAdd(&g[r * NMAPS + lane], c[r]);                 // ds_add_f32
    } else if (lane >= 24) {
        #pragma unroll
        for (int r = 0; r < 8; ++r)
            atomicAdd(&g[NMAPS * NMAPS + r * NMAPS + (lane - 24)], c[r]);
    }
    __syncthreads();

    // 128 global atomics per block: ws[(i*8+j)*32 + b] += g[blk][i*8+j]
    if (threadIdx.x < 2 * NMAPS * NMAPS) {
        int blk = threadIdx.x >> 6;          // 0 -> b0, 1 -> b1
        int t   = threadIdx.x & 63;          // i*8 + j
        atomicAdd(&ws[t * BATCH + (b0 + blk)], g[threadIdx.x]);
    }
}

// ---------------------------------------------------------------------------
// Finalize: ws[i][j][b] holds dots. Compute mean over (i<j, b) of
// |dot_ij| / max(sqrt(dot_ii)*sqrt(dot_jj), eps), write scalar.
// ---------------------------------------------------------------------------
__global__ __launch_bounds__(256) void finalize_kernel(
        const float* __restrict__ ws, float* __restrict__ out) {
    __shared__ float red[256];
    const int NITEMS = 28 * BATCH;   // 28 pairs x 32 batch elements
    float acc = 0.0f;
    for (int item = threadIdx.x; item < NITEMS; item += 256) {
        int pidx = item >> 5;
        int bb   = item & 31;
        // map pair index -> (i < j), row-major upper triangle
        int i = 0, rem = pidx;
        while (rem >= 7 - i) { rem -= 7 - i; ++i; }
        int j = i + 1 + rem;
        float dij = ws[(i * NMAPS + j) * BATCH + bb];
        float dii = ws[(i * NMAPS + i) * BATCH + bb];
        float djj = ws[(j * NMAPS + j) * BATCH + bb];
        float denom = fmaxf(sqrtf(dii) * sqrtf(djj), EPS);
        acc += fabsf(dij) / denom;
    }
    red[threadIdx.x] = acc;
    __syncthreads();
    for (int s = 128; s > 0; s >>= 1) {
        if (threadIdx.x < s) red[threadIdx.x] += red[threadIdx.x + s];
        __syncthreads();
    }
    if (threadIdx.x == 0) out[0] = red[0] / (float)NITEMS;
}

extern "C" void kernel_launch(void* const* d_in, const int* in_sizes, int n_in,
                              void* d_out, int out_size, void* d_ws, size_t ws_size,
                              hipStream_t stream) {
    const float* x = (const float*)d_in[0];
    float* out = (float*)d_out;
    float* ws  = (float*)d_ws;   // 8*8*32 floats = 8 KB

    zero_ws_kernel<<<(NMAPS * NMAPS * BATCH + 255) / 256, 256, 0, stream>>>(ws);

    // 16 pairs x 12 chunk-groups = 192 blocks, 1536 waves
    gram_wmma_kernel<<<NPAIRS * CGROUPS, 256, 0, stream>>>(x, ws);

    finalize_kernel<<<1, 256, 0, stream>>>(ws, out);
}